// MPC_73203422593503
// MI455X (gfx1250) — compile-verified
//
#include <hip/hip_runtime.h>

// ---------------------------------------------------------------------------
// Batched time-varying LQR (Riccati backward pass + rollout) for MI455X.
// T=64 timesteps (sequential), B=128 batch (parallel, 1 workgroup each),
// NS=64 states, NC=32 controls, NSC=96.
// All GEMMs use V_WMMA_F32_16X16X4_F32 (exact f32 matrix pipe, wave32).
// Each wave computes paired N-tiles so one A-operand load feeds two WMMAs.
// ---------------------------------------------------------------------------

typedef float v2f __attribute__((ext_vector_type(2)));
typedef float v8f __attribute__((ext_vector_type(8)));

constexpr int Tn  = 64;
constexpr int Bn  = 128;
constexpr int NS  = 64;
constexpr int NC  = 32;
constexpr int NSC = 96;

// LDS pads chosen for bank-conflict-free WMMA operand access (64 banks):
constexpr int VP = 65;   // V stride: column reads -> banks (row+k)%64, all distinct
constexpr int FP = 100;  // F/W/Q stride: 16-row column reads hit 16 distinct banks; 400B row, 8B-aligned
constexpr int KP = 68;   // K stride in forward pass

// LDS layout (floats)
constexpr int LDS_V   = 0;                  // 64*65
constexpr int LDS_F   = LDS_V + NS * VP;    // 64*100
constexpr int LDS_W   = LDS_F + NS * FP;    // 64*100  (reused: T1, K)
constexpr int LDS_Q   = LDS_W + NS * FP;    // 96*100  (col 96 holds q)
constexpr int LDS_TMP = LDS_Q + NSC * FP;   // 64
constexpr int LDS_vv  = LDS_TMP + NS;       // 64
constexpr int LDS_ff  = LDS_vv + NS;        // 64
constexpr int LDS_cc  = LDS_ff + NS;        // 96
constexpr int LDS_x   = LDS_cc + NSC;       // 64
constexpr int LDS_u   = LDS_x + NS;         // 32
constexpr int LDS_xn  = LDS_u + NC;         // 64
constexpr int LDS_kk  = LDS_xn + NS;        // 32
constexpr int LDS_TOTAL = LDS_kk + NC;      // 27040 floats = 108160 B (< 320KB WGP LDS)

__global__ __launch_bounds__(256) void lqr_mi455x_kernel(
    const float* __restrict__ Cg_all, const float* __restrict__ cg_all,
    const float* __restrict__ Fg_all, const float* __restrict__ fg_all,
    const float* __restrict__ xinit,  float* __restrict__ out,
    float* __restrict__ Ks, float* __restrict__ ksv)
{
    extern __shared__ float sm[];
    float* Vl  = sm + LDS_V;
    float* Fl  = sm + LDS_F;
    float* Wl  = sm + LDS_W;
    float* Ql  = sm + LDS_Q;
    float* tmp = sm + LDS_TMP;
    float* vl  = sm + LDS_vv;
    float* fl  = sm + LDS_ff;
    float* cl  = sm + LDS_cc;
    float* xl  = sm + LDS_x;
    float* ul  = sm + LDS_u;
    float* xnl = sm + LDS_xn;
    float* kl  = sm + LDS_kk;

    const int tid  = threadIdx.x;
    const int lane = tid & 31;
    const int wv   = tid >> 5;       // wave id 0..7
    const int l15  = lane & 15;
    const int lh   = lane >> 4;      // half-wave select (K split / M split)
    const int b    = blockIdx.x;

    // init carry: V = 0, v = 0
    for (int i = tid; i < NS * VP; i += 256) Vl[i] = 0.0f;
    if (tid < NS) vl[tid] = 0.0f;
    __syncthreads();

    // ======================= backward Riccati pass =======================
    for (int t = Tn - 1; t >= 0; --t) {
        const size_t tb = (size_t)t * Bn + b;
        const float* Fg = Fg_all + tb * (NS * NSC);
        const float* Cg = Cg_all + tb * (NSC * NSC);
        const float* cg = cg_all + tb * NSC;
        const float* fg = fg_all + tb * NS;

        // ---- stage F (float2), f, c into LDS; prefetch next timestep ----
        {
            const float2* Fg2 = reinterpret_cast<const float2*>(Fg);
            for (int i = tid; i < NS * (NSC / 2); i += 256) {
                const int r = i / (NSC / 2), cc = i % (NSC / 2);
                *reinterpret_cast<float2*>(&Fl[r * FP + 2 * cc]) = Fg2[i];
            }
        }
        if (tid < NS)  fl[tid] = fg[tid];
        if (tid < NSC) cl[tid] = cg[tid];
        if (t > 0) {
            const float* Cn = Cg - (size_t)Bn * (NSC * NSC);
            const float* Fn = Fg - (size_t)Bn * (NS * NSC);
            for (int i = tid; i < 288; i += 256)
                __builtin_prefetch(Cn + i * 32, 0, 1);   // 128B stride, covers 96*96
            if (tid < 192)
                __builtin_prefetch(Fn + tid * 32, 0, 1); // covers 64*96
        }
        __syncthreads();

        // ---- tmp = v + V f (threads 0..63) ; W = V * F (WMMA, 4x3 units) ----
        if (tid < NS) {
            float s = vl[tid];
            for (int j = 0; j < NS; ++j) s += Vl[tid * VP + j] * fl[j];
            tmp[tid] = s;
        }
        for (int uu = wv; uu < 12; uu += 8) {
            const int mb = (uu / 3) * 16, nb = (uu % 3) * 32;
            v8f acc0 = {}, acc1 = {};
#pragma unroll
            for (int kb = 0; kb < NS; kb += 4) {
                const int kr = kb + 2 * lh;
                v2f a, b0, b1;
                a.x  = Vl[(mb + l15) * VP + kr];
                a.y  = Vl[(mb + l15) * VP + kr + 1];
                b0.x = Fl[kr * FP + nb + l15];
                b0.y = Fl[(kr + 1) * FP + nb + l15];
                b1.x = Fl[kr * FP + nb + 16 + l15];
                b1.y = Fl[(kr + 1) * FP + nb + 16 + l15];
                acc0 = __builtin_amdgcn_wmma_f32_16x16x4_f32(
                    false, a, false, b0, (short)0, acc0, false, false);
                acc1 = __builtin_amdgcn_wmma_f32_16x16x4_f32(
                    false, a, false, b1, (short)0, acc1, false, false);
            }
#pragma unroll
            for (int r = 0; r < 8; ++r) {
                Wl[(mb + r + 8 * lh) * FP + nb + l15]      = acc0[r];
                Wl[(mb + r + 8 * lh) * FP + nb + 16 + l15] = acc1[r];
            }
        }
        __syncthreads();

        // ---- q = c + F^T tmp -> Q col 96 ; Q = C + F^T W (WMMA, 6x3 units) ----
        if (tid < NSC) {
            float s = cl[tid];
            for (int j = 0; j < NS; ++j) s += Fl[j * FP + tid] * tmp[j];
            Ql[tid * FP + NSC] = s;
        }
        for (int uu = wv; uu < 18; uu += 8) {
            const int mb = (uu / 3) * 16, nb = (uu % 3) * 32;
            v8f acc0, acc1;
#pragma unroll
            for (int r = 0; r < 8; ++r) {
                acc0[r] = Cg[(mb + r + 8 * lh) * NSC + nb + l15];
                acc1[r] = Cg[(mb + r + 8 * lh) * NSC + nb + 16 + l15];
            }
#pragma unroll
            for (int kb = 0; kb < NS; kb += 4) {
                const int kr = kb + 2 * lh;
                v2f a, b0, b1;
                a.x  = Fl[kr * FP + mb + l15];        // A[i][s] = F[s][i]
                a.y  = Fl[(kr + 1) * FP + mb + l15];
                b0.x = Wl[kr * FP + nb + l15];
                b0.y = Wl[(kr + 1) * FP + nb + l15];
                b1.x = Wl[kr * FP + nb + 16 + l15];
                b1.y = Wl[(kr + 1) * FP + nb + 16 + l15];
                acc0 = __builtin_amdgcn_wmma_f32_16x16x4_f32(
                    false, a, false, b0, (short)0, acc0, false, false);
                acc1 = __builtin_amdgcn_wmma_f32_16x16x4_f32(
                    false, a, false, b1, (short)0, acc1, false, false);
            }
#pragma unroll
            for (int r = 0; r < 8; ++r) {
                Ql[(mb + r + 8 * lh) * FP + nb + l15]      = acc0[r];
                Ql[(mb + r + 8 * lh) * FP + nb + 16 + l15] = acc1[r];
            }
        }
        __syncthreads();

        // ---- Gauss-Jordan on rows 64..95 of [Qux | Quu | q] (SPD, no pivoting).
        // Pivot column is skipped everywhere -> no read/write race on the pivot.
        for (int p = 0; p < NC; ++p) {
            const int rp = NS + p, cp = NS + p;
            const float pinv = 1.0f / Ql[rp * FP + cp];
            if (tid <= NSC && tid != cp) Ql[rp * FP + tid] *= pinv;
            __syncthreads();
            for (int idx = tid; idx < NC * (NSC + 1); idx += 256) {
                const int r = NS + idx / (NSC + 1);
                const int ccol = idx % (NSC + 1);
                if (r != rp && ccol != cp)
                    Ql[r * FP + ccol] -= Ql[r * FP + cp] * Ql[rp * FP + ccol];
            }
            __syncthreads();
        }
        // now rows 64..95: cols 0..63 = Quu^{-1}Qux, col 96 = Quu^{-1}qu

        // ---- spill K = -Quu^{-1}Qux, k = -Quu^{-1}qu ; vn ; T1 = Qxx + Qxu*K ----
        for (int idx = tid; idx < NC * NS; idx += 256)
            Ks[tb * (NC * NS) + idx] = -Ql[(NS + (idx >> 6)) * FP + (idx & 63)];
        if (tid < NC) ksv[tb * NC + tid] = -Ql[(NS + tid) * FP + NSC];
        if (tid < NS) {
            float s = Ql[tid * FP + NSC];              // qx
            for (int m = 0; m < NC; ++m)               // + Qxu * k  (k = -sol)
                s -= Ql[tid * FP + NS + m] * Ql[(NS + m) * FP + NSC];
            vl[tid] = s;
        }
        {   // 4x2 units (one per wave), each = two 16x16 tiles of 64x64
            const int mb = (wv >> 1) * 16, nb = (wv & 1) * 32;
            v8f acc0, acc1;
#pragma unroll
            for (int r = 0; r < 8; ++r) {              // init with Qxx tiles
                acc0[r] = Ql[(mb + r + 8 * lh) * FP + nb + l15];
                acc1[r] = Ql[(mb + r + 8 * lh) * FP + nb + 16 + l15];
            }
#pragma unroll
            for (int kb = 0; kb < NC; kb += 4) {
                const int kr = kb + 2 * lh;
                v2f a, b0, b1;
                a.x  = Ql[(mb + l15) * FP + NS + kr];       // Qxu
                a.y  = Ql[(mb + l15) * FP + NS + kr + 1];
                b0.x = -Ql[(NS + kr) * FP + nb + l15];      // K = -sol
                b0.y = -Ql[(NS + kr + 1) * FP + nb + l15];
                b1.x = -Ql[(NS + kr) * FP + nb + 16 + l15];
                b1.y = -Ql[(NS + kr + 1) * FP + nb + 16 + l15];
                acc0 = __builtin_amdgcn_wmma_f32_16x16x4_f32(
                    false, a, false, b0, (short)0, acc0, false, false);
                acc1 = __builtin_amdgcn_wmma_f32_16x16x4_f32(
                    false, a, false, b1, (short)0, acc1, false, false);
            }
#pragma unroll
            for (int r = 0; r < 8; ++r) {              // T1 into W buffer
                Wl[(mb + r + 8 * lh) * FP + nb + l15]      = acc0[r];
                Wl[(mb + r + 8 * lh) * FP + nb + 16 + l15] = acc1[r];
            }
        }
        __syncthreads();

        // ---- V = 0.5*(T1 + T1^T) ----
        for (int idx = tid; idx < NS * NS; idx += 256) {
            const int i = idx >> 6, j = idx & 63;
            Vl[i * VP + j] = 0.5f * (Wl[i * FP + j] + Wl[j * FP + i]);
        }
        __syncthreads();
    }

    // =========================== forward rollout ===========================
    if (tid < NS) xl[tid] = xinit[(size_t)b * NS + tid];
    __syncthreads();
    for (int t = 0; t < Tn; ++t) {
        const size_t tb = (size_t)t * Bn + b;
        const float* Kg = Ks + tb * (NC * NS);
        const float* Fg = Fg_all + tb * (NS * NSC);
        const float* fg = fg_all + tb * NS;
        {
            const float2* Kg2 = reinterpret_cast<const float2*>(Kg);
            for (int i = tid; i < NC * (NS / 2); i += 256) {
                const int r = i / (NS / 2), cc = i % (NS / 2);
                *reinterpret_cast<float2*>(&Wl[r * KP + 2 * cc]) = Kg2[i];
            }
            const float2* Fg2 = reinterpret_cast<const float2*>(Fg);
            for (int i = tid; i < NS * (NSC / 2); i += 256) {
                const int r = i / (NSC / 2), cc = i % (NSC / 2);
                *reinterpret_cast<float2*>(&Fl[r * FP + 2 * cc]) = Fg2[i];
            }
        }
        if (tid < NS) fl[tid] = fg[tid];
        if (tid < NC) kl[tid] = ksv[tb * NC + tid];
        __syncthreads();

        float* og = out + tb * NSC;
        if (tid < NC) {                                 // u = K x + k
            float s = kl[tid];
            for (int i = 0; i < NS; ++i) s += Wl[tid * KP + i] * xl[i];
            ul[tid] = s;
            og[NS + tid] = s;
        }
        if (tid < NS) og[tid] = xl[tid];                // tau = [x; u]
        __syncthreads();

        if (tid < NS) {                                 // x' = F tau + f
            float s = fl[tid];
            for (int i = 0; i < NS; ++i) s += Fl[tid * FP + i] * xl[i];
            for (int m = 0; m < NC; ++m) s += Fl[tid * FP + NS + m] * ul[m];
            xnl[tid] = s;
        }
        __syncthreads();
        if (tid < NS) xl[tid] = xnl[tid];
        __syncthreads();
    }
}

extern "C" void kernel_launch(void* const* d_in, const int* in_sizes, int n_in,
                              void* d_out, int out_size, void* d_ws, size_t ws_size,
                              hipStream_t stream) {
    const float* C  = (const float*)d_in[0];
    const float* c  = (const float*)d_in[1];
    const float* F  = (const float*)d_in[2];
    const float* f  = (const float*)d_in[3];
    const float* x0 = (const float*)d_in[4];
    float* out = (float*)d_out;

    // workspace: K gains [T,B,NC,NS] then k [T,B,NC]
    float* Ks  = (float*)d_ws;
    float* ksv = Ks + (size_t)Tn * Bn * NC * NS;

    const size_t smem = (size_t)LDS_TOTAL * sizeof(float);
    hipLaunchKernelGGL(lqr_mi455x_kernel, dim3(Bn), dim3(256), smem, stream,
                       C, c, F, f, x0, out, Ks, ksv);
}